// GraphAttentionLayer_74947179315827
// MI455X (gfx1250) — compile-verified
//
#include <hip/hip_runtime.h>
#include <math.h>

// GraphAttentionLayer for MI455X (gfx1250), fp32 WMMA path, v2.
//
//   0. zero d_out (h_new accum) and sum_w
//   1. bdst[k]   = sum_j a2[j] * W_dst[j][k]            (collapses dst GEMM)
//   2. s_dst[i]  = dot(target_vecs[i], bdst)
//   3. h_src_all = source @ W_srcT (WMMA), s_src epilogue
//   4. FUSED: 32-edge wave tile of h_rating via WMMA, s_rat epilogue,
//      attention + h_src gather + atomic segment-sum into d_out / sum_w
//   5. finalize: elu(h_new / (sum_w + 1e-8)) in place
//
// v2 changes: pair-interleaved LDS layout for W^T so B fragments are single
// conflict-free ds_load_b64 into aligned VGPR pairs (kills the v_dual_mov
// storm), float2 A loads, and 32-row M-tiles per wave (2x WMMA per ds load).

typedef float v2f __attribute__((ext_vector_type(2)));
typedef float v8f __attribute__((ext_vector_type(8)));

#define DIM 128
// W^T pair layout: lds[(k>>1)*GSTRIDE + 2*n + (k&1)]
// GSTRIDE = 288 words -> half-wave k-offset (+2 rows) lands +32 banks away.
#define GSTRIDE 288
#define GSTRIDE_F2 144

static constexpr int kNSrc = 100000;
static constexpr int kNDst = 50000;
static constexpr int kNEdge = 640000;

__device__ __forceinline__ float half16_reduce(float p) {
    p += __shfl_xor(p, 1, 32);
    p += __shfl_xor(p, 2, 32);
    p += __shfl_xor(p, 4, 32);
    p += __shfl_xor(p, 8, 32);
    return p;   // row sum replicated in all 16 lanes of the half-wave
}

__device__ __forceinline__ float wave32_reduce(float p) {
    p += __shfl_xor(p, 1, 32);
    p += __shfl_xor(p, 2, 32);
    p += __shfl_xor(p, 4, 32);
    p += __shfl_xor(p, 8, 32);
    p += __shfl_xor(p, 16, 32);
    return p;
}

__global__ void zero_kernel(float* __restrict__ a, int n) {
    int i = blockIdx.x * 256 + threadIdx.x;
    if (i < n) a[i] = 0.0f;
}

// bdst[k] = sum_j a2[j] * W_dst[j][k]  (a2 = a_w[128:256]); one 128-thread block
__global__ void bdst_kernel(const float* __restrict__ W_dst,
                            const float* __restrict__ a_w,
                            float* __restrict__ bdst) {
    int k = threadIdx.x;
    float acc = 0.0f;
    for (int j = 0; j < DIM; ++j)
        acc += a_w[DIM + j] * W_dst[j * DIM + k];
    bdst[k] = acc;
}

// s_dst[row] = dot(target_vecs[row], bdst); one wave per row
__global__ __launch_bounds__(256)
void sdst_kernel(const float* __restrict__ tv,
                 const float* __restrict__ bdst,
                 float* __restrict__ s_dst) {
    int gid  = blockIdx.x * 256 + threadIdx.x;
    int row  = gid >> 5;
    int lane = threadIdx.x & 31;
    if (row >= kNDst) return;
    const float4* r = (const float4*)(tv + (size_t)row * DIM);
    const float4* b = (const float4*)bdst;
    float4 x = r[lane];
    float4 y = b[lane];
    float p = x.x * y.x + x.y * y.y + x.z * y.z + x.w * y.w;
    p = wave32_reduce(p);
    if (lane == 0) s_dst[row] = p;
}

// MODE 0: h_out = X @ W^T, s_out = h_out @ avec      (source pass)
// MODE 1: fused rating GEMM + attention + segment-sum (edge pass)
// One wave handles a 32-row tile (two 16-row WMMA sub-tiles).
template <int MODE>
__global__ __launch_bounds__(256)
void gat_gemm(const float* __restrict__ X, const float* __restrict__ W,
              const float* __restrict__ avec,
              float* __restrict__ h_out, float* __restrict__ s_out,
              const int* __restrict__ src_idx, const int* __restrict__ dst_idx,
              const float* __restrict__ h_src_all,
              const float* __restrict__ s_src, const float* __restrict__ s_dst,
              float* __restrict__ h_new, float* __restrict__ sum_w,
              int ntiles) {
    __shared__ float ldsB[(DIM / 2) * GSTRIDE];   // 73.7 KB, pair-interleaved W^T

    int tid = threadIdx.x;
    // Stage W^T. W is row-major W[n][k]; read coalesced in k, write pair layout.
    for (int idx = tid; idx < DIM * DIM; idx += 256) {
        int n = idx >> 7;      // output column
        int k = idx & 127;     // K index
        ldsB[(k >> 1) * GSTRIDE + 2 * n + (k & 1)] = W[idx];
    }
    __syncthreads();

    int wave = tid >> 5;
    int lane = tid & 31;
    int tile = blockIdx.x * 8 + wave;
    if (tile >= ntiles) return;

    int m0    = tile * 32;
    int row_l = lane & 15;
    int hi    = lane >> 4;          // 0: K+{0,1}, 1: K+{2,3}
    const float2* Arow0 = (const float2*)(X + (size_t)(m0 + row_l) * DIM);
    const float2* Arow1 = (const float2*)(X + (size_t)(m0 + 16 + row_l) * DIM);
    const float2* ldsB2 = (const float2*)ldsB;

    v8f acc[2][8] = {};   // acc[t][nt][r] = h[m0+16t + r + 8*hi][nt*16+row_l]

    #pragma unroll 2
    for (int ks = 0; ks < DIM; ks += 4) {
        int kp = (ks >> 1) + hi;            // pair-group index for this half-wave
        float2 a0 = Arow0[kp];
        float2 a1 = Arow1[kp];
        v2f af0; af0[0] = a0.x; af0[1] = a0.y;
        v2f af1; af1[0] = a1.x; af1[1] = a1.y;
        const float2* bg = ldsB2 + (size_t)kp * GSTRIDE_F2;
        #pragma unroll
        for (int nt = 0; nt < 8; ++nt) {
            float2 b = bg[nt * 16 + row_l];   // one ds_load_b64, conflict-free
            v2f bf; bf[0] = b.x; bf[1] = b.y;
            acc[0][nt] = __builtin_amdgcn_wmma_f32_16x16x4_f32(
                false, af0, false, bf, (short)0, acc[0][nt], false, false);
            acc[1][nt] = __builtin_amdgcn_wmma_f32_16x16x4_f32(
                false, af1, false, bf, (short)0, acc[1][nt], false, false);
        }
    }

    // Epilogue: per-row dot with avec, reduced across each 16-lane half.
    float av[8];
    #pragma unroll
    for (int nt = 0; nt < 8; ++nt) av[nt] = avec[nt * 16 + row_l];

    int rbase = hi ? 8 : 0;
    #pragma unroll
    for (int t = 0; t < 2; ++t) {
        int mt = m0 + 16 * t;
        float sred[8];
        #pragma unroll
        for (int m = 0; m < 8; ++m) {
            float p = 0.0f;
            #pragma unroll
            for (int nt = 0; nt < 8; ++nt) p += acc[t][nt][m] * av[nt];
            sred[m] = half16_reduce(p);
        }

        if (MODE == 0) {
            #pragma unroll
            for (int m = 0; m < 8; ++m) {
                int row = mt + m + rbase;
                #pragma unroll
                for (int nt = 0; nt < 8; ++nt)
                    h_out[(size_t)row * DIM + nt * 16 + row_l] = acc[t][nt][m];
                if (row_l == 0) s_out[row] = sred[m];
            }
        } else {
            // lanes 0-15 handle edge mt+m, lanes 16-31 edge mt+m+8
            #pragma unroll
            for (int m = 0; m < 8; ++m) {
                int e  = mt + m + rbase;
                int si = src_idx[e];
                int di = dst_idx[e];
                float score = s_src[si] + s_dst[di] + sred[m];
                score = (score >= 0.0f) ? score : 0.2f * score;  // leaky_relu 0.2
                float att = expf(score);
                const float* hs = h_src_all + (size_t)si * DIM;
                float*       hn = h_new + (size_t)di * DIM;
                #pragma unroll
                for (int nt = 0; nt < 8; ++nt) {
                    float msg = acc[t][nt][m] + hs[nt * 16 + row_l];
                    atomicAdd(&hn[nt * 16 + row_l], att * msg);
                }
                if (row_l == 0) atomicAdd(&sum_w[di], att);
            }
        }
    }
}

__global__ void finalize_kernel(float* __restrict__ h_new,
                                const float* __restrict__ sum_w, int n) {
    int i = blockIdx.x * 256 + threadIdx.x;
    if (i >= n) return;
    float w = sum_w[i >> 7] + 1e-8f;
    float v = h_new[i] / w;
    h_new[i] = (v > 0.0f) ? v : expm1f(v);   // elu, alpha = 1
}

extern "C" void kernel_launch(void* const* d_in, const int* in_sizes, int n_in,
                              void* d_out, int out_size, void* d_ws, size_t ws_size,
                              hipStream_t stream) {
    const float* source_vecs = (const float*)d_in[0];
    const int*   edge_index  = (const int*)d_in[1];
    const float* target_vecs = (const float*)d_in[2];
    const float* rating_vecs = (const float*)d_in[3];
    const float* W_src       = (const float*)d_in[4];
    const float* W_dst       = (const float*)d_in[5];
    const float* W_rating    = (const float*)d_in[6];
    const float* a_w         = (const float*)d_in[7];
    float* out = (float*)d_out;

    // Workspace layout (floats): ~52 MB total
    float* ws        = (float*)d_ws;
    float* h_src_all = ws;                                  // N_SRC * 128
    float* s_src     = h_src_all + (size_t)kNSrc * DIM;     // N_SRC
    float* s_dst     = s_src + kNSrc;                       // N_DST
    float* sum_w     = s_dst + kNDst;                       // N_DST
    float* bdst      = sum_w + kNDst;                       // 128

    const int* src_idx = edge_index;            // edge_index[0, :]
    const int* dst_idx = edge_index + kNEdge;   // edge_index[1, :]

    // 0. zero accumulators (d_out is the h_new accumulator)
    zero_kernel<<<(kNDst * DIM + 255) / 256, 256, 0, stream>>>(out, kNDst * DIM);
    zero_kernel<<<(kNDst + 255) / 256, 256, 0, stream>>>(sum_w, kNDst);

    // 1-2. collapsed dst path
    bdst_kernel<<<1, 128, 0, stream>>>(W_dst, a_w, bdst);
    sdst_kernel<<<(kNDst * 32 + 255) / 256, 256, 0, stream>>>(target_vecs, bdst, s_dst);

    // 3. source GEMM + s_src epilogue (32-row wave tiles)
    int t_src = kNSrc / 32;   // 3125
    gat_gemm<0><<<(t_src + 7) / 8, 256, 0, stream>>>(
        source_vecs, W_src, a_w /* a1 */, h_src_all, s_src,
        nullptr, nullptr, nullptr, nullptr, nullptr, nullptr, nullptr, t_src);

    // 4. fused rating GEMM + attention + segment-sum
    int t_e = kNEdge / 32;    // 20000
    gat_gemm<1><<<(t_e + 7) / 8, 256, 0, stream>>>(
        rating_vecs, W_rating, a_w + 2 * DIM /* a3 */, nullptr, nullptr,
        src_idx, dst_idx, h_src_all, s_src, s_dst, out, sum_w, t_e);

    // 5. normalize + ELU
    finalize_kernel<<<(kNDst * DIM + 255) / 256, 256, 0, stream>>>(out, sum_w, kNDst * DIM);
}